// HungarianMatcher_54760833024710
// MI455X (gfx1250) — compile-verified
//
#include <hip/hip_runtime.h>

// Problem constants (from reference)
#define BS 16
#define NQ 900
#define NC 91
#define NTOT (BS * NQ)      // 14400
#define TT 1024             // targets
#define ROWS 16             // n-rows per workgroup  -> 900 blocks

#define C_CLASS 2.0f
#define C_BBOX  5.0f
#define C_GIOU  2.0f
#define C_INNER 9999.0f
#define F_ALPHA 0.25f
#define F_EPS   1e-8f

typedef float v4f __attribute__((ext_vector_type(4)));

__device__ __forceinline__ float fast_rcp(float x) { return __builtin_amdgcn_rcpf(x); }
__device__ __forceinline__ float sigmoid_f(float x) {
    return fast_rcp(1.0f + __expf(-x));
}

// ---------------------------------------------------------------------------
// Prep: build target table in d_ws, 8 floats per target (32B rows):
//   [x0, y0, x1, y1, area, label(bits), 0, 0]
// ---------------------------------------------------------------------------
__global__ void hm_prep_tgt(const float* __restrict__ tgt_boxes,
                            const int* __restrict__ tgt_labels,
                            float* __restrict__ ws_tgt) {
    int t = blockIdx.x * blockDim.x + threadIdx.x;
    if (t >= TT) return;
    float cx = tgt_boxes[t * 4 + 0];
    float cy = tgt_boxes[t * 4 + 1];
    float w  = tgt_boxes[t * 4 + 2];
    float h  = tgt_boxes[t * 4 + 3];
    float x0 = cx - 0.5f * w, y0 = cy - 0.5f * h;
    float x1 = cx + 0.5f * w, y1 = cy + 0.5f * h;
    v4f a = {x0, y0, x1, y1};
    v4f b = {(x1 - x0) * (y1 - y0), __int_as_float(tgt_labels[t]), 0.0f, 0.0f};
    v4f* dst = (v4f*)ws_tgt;
    dst[t * 2 + 0] = a;
    dst[t * 2 + 1] = b;
}

// ---------------------------------------------------------------------------
// Main: one block = 16 pred rows x all 1024 targets.
// Target table staged to LDS via CDNA5 async global->LDS DMA, overlapped
// with in-LDS construction of the per-row focal class-cost table.
// ---------------------------------------------------------------------------
__global__ __launch_bounds__(256) void hm_main(
    const float* __restrict__ pred_logits,   // [N, NC]
    const float* __restrict__ pred_boxes,    // [N, 4]  cxcywh
    const float* __restrict__ pred_points,   // [N, 2]
    const float* __restrict__ pred_objs,     // [N]
    const float* __restrict__ ws_tgt,        // [TT, 8]
    float* __restrict__ out)                 // [N, TT]
{
    __shared__ __align__(16) float s_tgt[TT * 8];     // 32768 B
    __shared__ __align__(16) float s_tab[ROWS * NC];  // 5824 B (pre-scaled by C_CLASS)
    __shared__ __align__(16) float s_row[ROWS * 8];   // x0,y0,x1,y1,area,px,py,objsig

    const int tid = threadIdx.x;
    const int n0  = blockIdx.x * ROWS;

    // ---- 1) kick off async global->LDS copy of the 32KB target table ------
    {
        unsigned lds_base = (unsigned)(unsigned long long)(&s_tgt[0]);
        const char* gbase = (const char*)ws_tgt;
#pragma unroll
        for (int i = 0; i < 8; ++i) {
            unsigned loff = lds_base + (unsigned)((i * 256 + tid) * 16);
            unsigned long long ga =
                (unsigned long long)(gbase + (size_t)(i * 256 + tid) * 16);
            asm volatile("global_load_async_to_lds_b128 %0, %1, off"
                         :: "v"(loff), "v"(ga) : "memory");
        }
    }

    // ---- 2) per-row pred data (incl. objectness sigmoid, computed once) ---
    if (tid < ROWS) {
        int n = n0 + tid;
        float cx = pred_boxes[n * 4 + 0];
        float cy = pred_boxes[n * 4 + 1];
        float w  = pred_boxes[n * 4 + 2];
        float h  = pred_boxes[n * 4 + 3];
        float x0 = cx - 0.5f * w, y0 = cy - 0.5f * h;
        float x1 = cx + 0.5f * w, y1 = cy + 0.5f * h;
        s_row[tid * 8 + 0] = x0;
        s_row[tid * 8 + 1] = y0;
        s_row[tid * 8 + 2] = x1;
        s_row[tid * 8 + 3] = y1;
        s_row[tid * 8 + 4] = (x1 - x0) * (y1 - y0);
        s_row[tid * 8 + 5] = pred_points[n * 2 + 0];
        s_row[tid * 8 + 6] = pred_points[n * 2 + 1];
        s_row[tid * 8 + 7] = sigmoid_f(pred_objs[n]);
    }
    __syncthreads();   // s_row visible (async DMA keeps running in background)

    // ---- 3) focal class-cost table for our 16 rows (overlaps async DMA) ---
    for (int e = tid; e < ROWS * NC; e += 256) {
        int r = e / NC, c = e - r * NC;
        int n = n0 + r;
        float p   = sigmoid_f(pred_logits[n * NC + c]) * s_row[r * 8 + 7];
        float omp = 1.0f - p;
        float pos = F_ALPHA * omp * omp * (-__logf(p + F_EPS));
        float neg = (1.0f - F_ALPHA) * p * p * (-__logf(omp + F_EPS));
        s_tab[e] = C_CLASS * (pos - neg);
    }

    // ---- 4) drain async DMA, make all LDS writes visible -------------------
    asm volatile("s_wait_asynccnt 0x0" ::: "memory");
    __syncthreads();

    // ---- 5) each thread owns 4 consecutive target columns ------------------
    float tx0[4], ty0[4], tx1[4], ty1[4], tar[4];
    int   tlab[4];
    const v4f* stv = (const v4f*)s_tgt;
#pragma unroll
    for (int c = 0; c < 4; ++c) {
        int t = tid * 4 + c;
        v4f a = stv[t * 2 + 0];
        v4f b = stv[t * 2 + 1];
        tx0[c] = a.x; ty0[c] = a.y; tx1[c] = a.z; ty1[c] = a.w;
        tar[c] = b.x;
        tlab[c] = __float_as_int(b.y);
    }

    // ---- 6) 16-row sweep ----------------------------------------------------
    const v4f* srv = (const v4f*)s_row;
    float* obase = out + (size_t)n0 * TT + (size_t)tid * 4;
    for (int r = 0; r < ROWS; ++r) {
        v4f ra = srv[r * 2 + 0];
        v4f rb = srv[r * 2 + 1];
        float x0 = ra.x, y0 = ra.y, x1 = ra.z, y1 = ra.w;
        float area1 = rb.x, px = rb.y, py = rb.z;
        const float* tabr = &s_tab[r * NC];

        float res[4];
#pragma unroll
        for (int c = 0; c < 4; ++c) {
            // L1 bbox cost on xyxy
            float bb = fabsf(x0 - tx0[c]) + fabsf(y0 - ty0[c]) +
                       fabsf(x1 - tx1[c]) + fabsf(y1 - ty1[c]);
            // intersection / union
            float ix0 = fmaxf(x0, tx0[c]), iy0 = fmaxf(y0, ty0[c]);
            float ix1 = fminf(x1, tx1[c]), iy1 = fminf(y1, ty1[c]);
            float iw = fmaxf(ix1 - ix0, 0.0f), ih = fmaxf(iy1 - iy0, 0.0f);
            float inter = iw * ih;
            float uni   = area1 + tar[c] - inter;
            // enclosing box
            float cx0 = fminf(x0, tx0[c]), cy0 = fminf(y0, ty0[c]);
            float cx1 = fmaxf(x1, tx1[c]), cy1 = fmaxf(y1, ty1[c]);
            float cw = fmaxf(cx1 - cx0, 0.0f), ch = fmaxf(cy1 - cy0, 0.0f);
            float areac = cw * ch;
            float iou = inter * fast_rcp(uni);
            float uoc = uni * fast_rcp(areac);
            // cost_giou = -giou = 1 - iou - union/area_c  (scaled by C_GIOU)
            // inner-point cost
            float md = fminf(fminf(px - tx0[c], py - ty0[c]),
                             fminf(tx1[c] - px, ty1[c] - py));
            float innerp = (md >= 0.0f) ? 0.0f : C_INNER;
            float cls = tabr[tlab[c]];
            res[c] = C_BBOX * bb + cls + C_GIOU * (1.0f - iou - uoc) + innerp;
        }
        v4f rv = {res[0], res[1], res[2], res[3]};
        __builtin_nontemporal_store(rv, (v4f*)(obase + (size_t)r * TT));
    }
}

// ---------------------------------------------------------------------------
extern "C" void kernel_launch(void* const* d_in, const int* in_sizes, int n_in,
                              void* d_out, int out_size, void* d_ws, size_t ws_size,
                              hipStream_t stream) {
    const float* pred_logits = (const float*)d_in[0];   // [16,900,91]
    const float* pred_boxes  = (const float*)d_in[1];   // [16,900,4]
    const float* pred_points = (const float*)d_in[2];   // [16,900,2]
    const float* pred_objs   = (const float*)d_in[3];   // [16,900,1]
    const int*   tgt_labels  = (const int*)d_in[4];     // [1024]
    const float* tgt_boxes   = (const float*)d_in[5];   // [1024,4]

    float* ws_tgt = (float*)d_ws;                       // 32 KB used

    hm_prep_tgt<<<TT / 256, 256, 0, stream>>>(tgt_boxes, tgt_labels, ws_tgt);
    hm_main<<<NTOT / ROWS, 256, 0, stream>>>(pred_logits, pred_boxes, pred_points,
                                             pred_objs, ws_tgt, (float*)d_out);
}